// MinibatchDiscrimination_70806830842566
// MI455X (gfx1250) — compile-verified
//
#include <hip/hip_runtime.h>
#include <hip/hip_bf16.h>

// Shapes from the reference
#define BATCH 256
#define IN_F 1024
#define OUT_F 128
#define KD 16
#define NCOL (OUT_F * KD)      // 2048
#define OUT_ROW (IN_F + OUT_F) // 1152

typedef __attribute__((ext_vector_type(2))) float v2f;
typedef __attribute__((ext_vector_type(8))) float v8f;

// ---------------------------------------------------------------------------
// Kernel 1: M = x @ W  (x:[256,1024], W:[1024,2048] row-major, M:[256,2048])
// FP32 WMMA 16x16x4. Each wave owns a 16(M) x 64(N) tile -> 4 accumulators.
// 512 wave-tiles total = 64 blocks * 8 waves.
// ---------------------------------------------------------------------------
__global__ __launch_bounds__(256) void mbd_gemm_wmma(
    const float* __restrict__ x, const float* __restrict__ W,
    float* __restrict__ M) {
  const int lane  = threadIdx.x & 31;
  const int wave  = threadIdx.x >> 5;
  const int wg    = blockIdx.x * 8 + wave;   // 0..511
  const int mt    = wg >> 5;                 // 0..15  (16-row tile)
  const int nt    = wg & 31;                 // 0..31  (64-col tile)

  const int row   = lane & 15;               // A row / B-D col within tile
  const int khalf = lane >> 4;               // which K pair this lane holds

  const int m0 = mt * 16;
  const int n0 = nt * 64;

  v8f acc0 = {}, acc1 = {}, acc2 = {}, acc3 = {};

  const float* xrow = x + (m0 + row) * IN_F + 2 * khalf;

  for (int kb = 0; kb < IN_F; kb += 4) {
    // A fragment: A[row][kb + 2*khalf + j], j=0,1 (contiguous -> b64 load)
    v2f a;
    a.x = xrow[kb + 0];
    a.y = xrow[kb + 1];

    const float* wbase = W + (kb + 2 * khalf) * NCOL + n0 + row;
    // 4 B fragments for columns n0 + t*16 + row
    v2f b0, b1, b2, b3;
    b0.x = wbase[0];         b0.y = wbase[NCOL];
    b1.x = wbase[16];        b1.y = wbase[NCOL + 16];
    b2.x = wbase[32];        b2.y = wbase[NCOL + 32];
    b3.x = wbase[48];        b3.y = wbase[NCOL + 48];

    acc0 = __builtin_amdgcn_wmma_f32_16x16x4_f32(false, a, false, b0,
                                                 (short)0, acc0, false, false);
    acc1 = __builtin_amdgcn_wmma_f32_16x16x4_f32(false, a, false, b1,
                                                 (short)0, acc1, false, false);
    acc2 = __builtin_amdgcn_wmma_f32_16x16x4_f32(false, a, false, b2,
                                                 (short)0, acc2, false, false);
    acc3 = __builtin_amdgcn_wmma_f32_16x16x4_f32(false, a, false, b3,
                                                 (short)0, acc3, false, false);
  }

  // D layout: vgpr r holds D[r + 8*khalf][col = row]
  float* mout = M + (m0 + 8 * khalf) * NCOL + n0 + row;
#pragma unroll
  for (int r = 0; r < 8; ++r) {
    mout[r * NCOL + 0]  = acc0[r];
    mout[r * NCOL + 16] = acc1[r];
    mout[r * NCOL + 32] = acc2[r];
    mout[r * NCOL + 48] = acc3[r];
  }
}

// ---------------------------------------------------------------------------
// Kernel 2: pairwise L1 + exp reduction, one block per output feature o.
// Stage M[:, o, :] (256 x 16 f32 = 16KB) in LDS; thread j keeps its own row
// in registers and sweeps all 256 partner rows (LDS broadcast reads).
// Writes o_b directly into the concat layout: out[j*1152 + 1024 + o].
// ---------------------------------------------------------------------------
__global__ __launch_bounds__(256) void mbd_pairwise(
    const float* __restrict__ M, float* __restrict__ out) {
  __shared__ float s[BATCH * KD];  // 16 KB
  const int o = blockIdx.x;
  const int j = threadIdx.x;

  // Thread j loads row i=j of the slab: M[j*2048 + o*16 + k], k=0..15
  const float4* src = (const float4*)(M + j * NCOL + o * KD);
  float4 r0 = src[0], r1 = src[1], r2 = src[2], r3 = src[3];
  float4* dst = (float4*)(s + j * KD);
  dst[0] = r0; dst[1] = r1; dst[2] = r2; dst[3] = r3;
  __syncthreads();

  float mj[KD];
  *(float4*)(mj + 0)  = r0;
  *(float4*)(mj + 4)  = r1;
  *(float4*)(mj + 8)  = r2;
  *(float4*)(mj + 12) = r3;

  float acc = 0.0f;
  for (int i = 0; i < BATCH; ++i) {
    const float* ri = s + i * KD;  // same address across wave -> broadcast
    float l1 = 0.0f;
#pragma unroll
    for (int k = 0; k < KD; ++k) l1 += fabsf(ri[k] - mj[k]);
    acc += __expf(-l1);
  }
  // subtract self term exp(0)=1
  out[j * OUT_ROW + IN_F + o] = acc - 1.0f;
}

// ---------------------------------------------------------------------------
// Kernel 3: copy x into out[:, :1024] (float4 vectorized, 256 rows).
// ---------------------------------------------------------------------------
__global__ __launch_bounds__(256) void mbd_copy_x(
    const float* __restrict__ x, float* __restrict__ out) {
  const int idx = blockIdx.x * blockDim.x + threadIdx.x;  // over 256*256 f4s
  const int r  = idx >> 8;         // row 0..255
  const int c4 = idx & 255;        // float4 index 0..255 (1024 floats)
  const float4 v = ((const float4*)(x + r * IN_F))[c4];
  ((float4*)(out + r * OUT_ROW))[c4] = v;
}

extern "C" void kernel_launch(void* const* d_in, const int* in_sizes, int n_in,
                              void* d_out, int out_size, void* d_ws,
                              size_t ws_size, hipStream_t stream) {
  const float* x = (const float*)d_in[0];            // [256, 1024]
  const float* T = (const float*)d_in[1];            // [1024, 128, 16] == W [1024,2048]
  float* out = (float*)d_out;                        // [256, 1152]
  float* M = (float*)d_ws;                           // [256, 2048] scratch (2 MB)

  (void)in_sizes; (void)n_in; (void)out_size; (void)ws_size;

  mbd_gemm_wmma<<<dim3(64), dim3(256), 0, stream>>>(x, T, M);
  mbd_pairwise<<<dim3(OUT_F), dim3(256), 0, stream>>>(M, out);
  mbd_copy_x<<<dim3(256), dim3(256), 0, stream>>>(x, out);
}